// InfoNCELoss_7103875907855
// MI455X (gfx1250) — compile-verified
//
#include <hip/hip_runtime.h>
#include <hip/hip_bf16.h>
#include <math.h>

// Problem constants (from the reference)
#define BSZ   32
#define TLEN  500
#define VOC   8192
#define TCHUNK 4          // timesteps per block in phase 1
#define NTB   (TLEN / TCHUNK)   // 125 blocks

typedef __attribute__((ext_vector_type(2))) float v2f;
typedef __attribute__((ext_vector_type(8))) float v8f;

// ---------------------------------------------------------------------------
// Phase 1: partial cross[i,j] accumulation over a chunk of timesteps.
// Thread (i = tid&31, j = tid>>5): each wave handles one j for all 32 i's,
// so the wave's 32 gathers fall inside one 32KB row x[j,t,:].
// Writes partials[block][j*32+i] (deterministic, no atomics).
// ---------------------------------------------------------------------------
__global__ __launch_bounds__(1024) void infonce_phase1(
    const float* __restrict__ x, const int* __restrict__ Y,
    float* __restrict__ partials)
{
    __shared__ int yIdx[TCHUNK][BSZ];   // [t_local][i]

    const int tid = threadIdx.x;
    const int i   = tid & 31;
    const int j   = tid >> 5;
    const int t0  = blockIdx.x * TCHUNK;

    // Load the Y indices this block needs (TCHUNK*BSZ ints)
    if (tid < TCHUNK * BSZ) {
        const int tl = tid & (TCHUNK - 1);
        const int ii = tid >> 2;          // TCHUNK == 4
        yIdx[tl][ii] = Y[ii * TLEN + t0 + tl];
    }
    __syncthreads();

    const float* xj = x + (size_t)j * TLEN * VOC;
    float acc = 0.0f;
#pragma unroll
    for (int tl = 0; tl < TCHUNK; ++tl) {
        const int y = yIdx[tl][i];
        float v = xj[(size_t)(t0 + tl) * VOC + y];
        v = fminf(fmaxf(v, -30.0f), 30.0f);   // clip before summing, as reference
        acc += v;
    }
    partials[(size_t)blockIdx.x * (BSZ * BSZ) + tid] = acc;
}

// ---------------------------------------------------------------------------
// Phase 2: fold partials -> cross (32x32), logsumexp per row using WMMA for
// the Sum_j exp(cross[i,j]-m_i) row reduction (E * ones via
// v_wmma_f32_16x16x4_f32), then the scalar loss/correct outputs.
// ---------------------------------------------------------------------------
__global__ __launch_bounds__(1024) void infonce_phase2(
    const float* __restrict__ partials, const float* __restrict__ Bbuf,
    float* __restrict__ out)
{
    __shared__ float crossLDS[BSZ * BSZ];   // [j*32 + i]
    __shared__ float mI[BSZ];
    __shared__ float numer[BSZ];
    __shared__ float eLDS[BSZ][BSZ + 1];    // [i][j], padded
    __shared__ float sLDS[BSZ];
    __shared__ float ltLDS[BSZ];

    const int tid = threadIdx.x;

    // Deterministic fold of the 125 partials for this (i,j) slot
    float c = 0.0f;
    for (int b = 0; b < NTB; ++b)
        c += partials[(size_t)b * (BSZ * BSZ) + tid];
    crossLDS[tid] = c;
    __syncthreads();

    const float B0 = Bbuf[0];

    // Per-row max (including B0, since denominator_ = logsumexp({row, B0}))
    if (tid < BSZ) {
        const int i = tid;
        float mx = B0;
        for (int j = 0; j < BSZ; ++j)
            mx = fmaxf(mx, crossLDS[j * BSZ + i]);
        mI[i]    = mx;
        numer[i] = crossLDS[i * BSZ + i];
    }
    __syncthreads();

    // E[i][j] = exp(cross[i][j] - m[i])
    {
        const int i = tid & 31;
        const int j = tid >> 5;
        eLDS[i][j] = expf(crossLDS[j * BSZ + i] - mI[i]);
    }
    __syncthreads();

    // Row sums of E via WMMA: D = A(16x4) * ones(4x16) + C, chained over the
    // 32 columns (8 chunks of K=4). Wave 0 -> rows 0..15, wave 1 -> rows 16..31.
    const int warp = tid >> 5;
    const int lane = tid & 31;
    if (warp < 2) {
        const int r0  = warp * 16;
        const int row = r0 + (lane & 15);         // A: M = lane % 16
        const int kb  = (lane >> 4) << 1;         // A vgpr0 K = 0 (lo) / 2 (hi)

        v2f ones; ones[0] = 1.0f; ones[1] = 1.0f; // B layout-invariant (all 1s)
        v8f acc = {};
#pragma unroll
        for (int c4 = 0; c4 < 8; ++c4) {
            v2f a;
            a[0] = eLDS[row][4 * c4 + kb];        // K = kb
            a[1] = eLDS[row][4 * c4 + kb + 1];    // K = kb + 1
            acc = __builtin_amdgcn_wmma_f32_16x16x4_f32(
                /*neg_a=*/false, a, /*neg_b=*/false, ones,
                /*c_mod=*/(short)0, acc, /*reuse_a=*/false, /*reuse_b=*/false);
        }
        // D: vgpr v, lane l -> M = v + 8*(l>=16), N = l%16. Column 0 lives in
        // lanes 0 (rows r0..r0+7) and 16 (rows r0+8..r0+15).
        if ((lane & 15) == 0) {
            const int base = r0 + ((lane >> 4) << 3);
#pragma unroll
            for (int v = 0; v < 8; ++v)
                sLDS[base + v] = acc[v];
        }
    }
    __syncthreads();

    // loss_terms[i] = numer[i] - (m[i] + log(rowsum + exp(B0 - m[i])))
    if (tid < BSZ) {
        const int i = tid;
        const float s   = sLDS[i] + expf(B0 - mI[i]);
        const float den = mI[i] + logf(s);
        ltLDS[i] = numer[i] - den;
    }
    __syncthreads();

    if (tid == 0) {
        float sl = 0.0f, sc = 0.0f;
        for (int i = 0; i < BSZ; ++i) {
            sl += ltLDS[i];
            sc += expf(ltLDS[i]);
        }
        out[0] = -sl / (float)(BSZ * TLEN);   // loss
        out[1] = sc * (float)TLEN;            // correct
    }
}

// ---------------------------------------------------------------------------
extern "C" void kernel_launch(void* const* d_in, const int* in_sizes, int n_in,
                              void* d_out, int out_size, void* d_ws, size_t ws_size,
                              hipStream_t stream) {
    (void)in_sizes; (void)n_in; (void)out_size; (void)ws_size;
    const float* x    = (const float*)d_in[0];
    const int*   Y    = (const int*)d_in[1];
    const float* Bbuf = (const float*)d_in[2];
    float*       out  = (float*)d_out;
    float*       partials = (float*)d_ws;   // NTB * 1024 floats = 512 KB

    infonce_phase1<<<NTB, 1024, 0, stream>>>(x, Y, partials);
    infonce_phase2<<<1, 1024, 0, stream>>>(partials, Bbuf, out);
}